// VoteLoss_new_2370821948147
// MI455X (gfx1250) — compile-verified
//
#include <hip/hip_runtime.h>
#include <math.h>

typedef __attribute__((ext_vector_type(2))) float v2f;
typedef __attribute__((ext_vector_type(8))) float v8f;

#define N_POS 8192
#define N_ANC 8192

// ---------------------------------------------------------------------------
// Prep: transform anchors, build packed WMMA operands.
//   A-pack (per point q): (-2qx, -2qy, -2qz, 1)        -> A matrix rows
//   B-pack (per point q): ( qx,  qy,  qz, |q|^2)       -> B matrix columns
//   sq    (per point q): |q|^2                          -> C accumulator init
// With these, WMMA D = A*B + C gives d2 = |p|^2 - 2 p.a + |a|^2 directly.
// ---------------------------------------------------------------------------
__global__ __launch_bounds__(256)
void prep_kernel(const float* __restrict__ pos, const float* __restrict__ anc,
                 const float* __restrict__ T,
                 float4* __restrict__ posA4, float4* __restrict__ posB4, float* __restrict__ sqPos,
                 float4* __restrict__ ancA4, float4* __restrict__ ancB4, float* __restrict__ sqAnc) {
    int i = blockIdx.x * blockDim.x + threadIdx.x;
    if (i >= N_POS) return;

    float px = pos[i * 3 + 0], py = pos[i * 3 + 1], pz = pos[i * 3 + 2];
    float sp = px * px + py * py + pz * pz;
    posA4[i] = make_float4(-2.f * px, -2.f * py, -2.f * pz, 1.f);
    posB4[i] = make_float4(px, py, pz, sp);
    sqPos[i] = sp;

    float ax = anc[i * 3 + 0], ay = anc[i * 3 + 1], az = anc[i * 3 + 2];
    // anc' = R a + t ; transform is row-major 4x4
    float tx = T[0] * ax + T[1] * ay + T[2]  * az + T[3];
    float ty = T[4] * ax + T[5] * ay + T[6]  * az + T[7];
    float tz = T[8] * ax + T[9] * ay + T[10] * az + T[11];
    float sa = tx * tx + ty * ty + tz * tz;
    ancA4[i] = make_float4(-2.f * tx, -2.f * ty, -2.f * tz, 1.f);
    ancB4[i] = make_float4(tx, ty, tz, sa);
    sqAnc[i] = sa;
}

// ---------------------------------------------------------------------------
// Chamfer min via V_WMMA_F32_16X16X4_F32.
// Each wave owns 16 "A" rows, loops over all B points in 16-wide tiles.
// A 16x4 layout (ISA 7.12.2): lanes 0-15 hold (K0,K1) of row M=lane,
// lanes 16-31 hold (K2,K3)  -> one float2 load per lane from packed float4.
// C/D layout: VGPR v -> row M = v + 8*(lane>=16), col N = lane&15.
// ---------------------------------------------------------------------------
__global__ __launch_bounds__(256)
void chamfer_min_kernel(const float4* __restrict__ Apk, const float* __restrict__ sqA,
                        const float4* __restrict__ Bpk, float* __restrict__ minOut,
                        int nB) {
    const int lane    = threadIdx.x & 31;
    const int wave    = blockIdx.x * 8 + (threadIdx.x >> 5);
    const int rowbase = wave * 16;
    const int hi      = lane >> 4;   // 0: lanes 0-15, 1: lanes 16-31
    const int lo      = lane & 15;

    const float2* A2 = reinterpret_cast<const float2*>(Apk);
    const float2* B2 = reinterpret_cast<const float2*>(Bpk);

    v2f a;
    {
        float2 ta = A2[(size_t)(rowbase + lo) * 2 + hi];
        a.x = ta.x; a.y = ta.y;
    }

    v8f cinit;
#pragma unroll
    for (int v = 0; v < 8; ++v)
        cinit[v] = sqA[rowbase + v + hi * 8];

    v8f mn;
#pragma unroll
    for (int v = 0; v < 8; ++v) mn[v] = 3.0e38f;

    for (int cb = 0; cb < nB; cb += 16) {
        float2 tb = B2[(size_t)(cb + lo) * 2 + hi];
        v2f b; b.x = tb.x; b.y = tb.y;
        // D = A*B + C  -> 16x16 tile of raw squared distances
        v8f d = __builtin_amdgcn_wmma_f32_16x16x4_f32(
            /*neg_a=*/false, a, /*neg_b=*/false, b,
            /*c_mod=*/(short)0, cinit, /*reuse_a=*/false, /*reuse_b=*/false);
#pragma unroll
        for (int v = 0; v < 8; ++v) mn[v] = fminf(mn[v], d[v]);
    }

    // Reduce across the N dimension (lanes within each 16-lane group).
#pragma unroll
    for (int m = 1; m < 16; m <<= 1) {
#pragma unroll
        for (int v = 0; v < 8; ++v)
            mn[v] = fminf(mn[v], __shfl_xor(mn[v], m, 32));
    }
    if (lo == 0) {
#pragma unroll
        for (int v = 0; v < 8; ++v)
            minOut[rowbase + v + hi * 8] = mn[v];   // lane0: rows v, lane16: rows v+8
    }
}

// ---------------------------------------------------------------------------
// One-pass mask reduction: 64 blocks x 128 rows each; every thread owns 32
// fixed columns (coalesced int4 strides), OR-accumulates column flags in
// registers, row flags via LDS. 256 MB read exactly once.
// ---------------------------------------------------------------------------
__global__ __launch_bounds__(256)
void mask_kernel(const int* __restrict__ mask, int* __restrict__ rowAny,
                 int* __restrict__ colAny) {
    __shared__ int rowflag[128];
    const int t = threadIdx.x;
    const int rowbase = blockIdx.x * 128;
    if (t < 128) rowflag[t] = 0;
    __syncthreads();

    int4 acc[8];
#pragma unroll
    for (int k = 0; k < 8; ++k) acc[k] = make_int4(0, 0, 0, 0);

    for (int r = 0; r < 128; ++r) {
        const int4* rp = reinterpret_cast<const int4*>(mask + (size_t)(rowbase + r) * N_ANC);
        int rowor = 0;
#pragma unroll
        for (int k = 0; k < 8; ++k) {
            int4 m = rp[t + k * 256];          // columns t*4 + k*1024 .. +3
            acc[k].x |= m.x; acc[k].y |= m.y; acc[k].z |= m.z; acc[k].w |= m.w;
            rowor |= m.x | m.y | m.z | m.w;
        }
        if (rowor) atomicOr(&rowflag[r], 1);
    }
    __syncthreads();
    if (t < 128) rowAny[rowbase + t] = rowflag[t] ? 1 : 0;

#pragma unroll
    for (int k = 0; k < 8; ++k) {
        int c = t * 4 + k * 1024;
        if (acc[k].x) atomicOr(&colAny[c + 0], 1);
        if (acc[k].y) atomicOr(&colAny[c + 1], 1);
        if (acc[k].z) atomicOr(&colAny[c + 2], 1);
        if (acc[k].w) atomicOr(&colAny[c + 3], 1);
    }
}

__global__ void zero_kernel(int* __restrict__ p, int n) {
    int i = blockIdx.x * blockDim.x + threadIdx.x;
    if (i < n) p[i] = 0;
}

// ---------------------------------------------------------------------------
// Final: sqrt(max(min_d2,0)) * anyflag, masked means, sum of both directions.
// ---------------------------------------------------------------------------
__global__ __launch_bounds__(256)
void loss_kernel(const float* __restrict__ minPos, const int* __restrict__ rowAny,
                 const float* __restrict__ minAnc, const int* __restrict__ colAny,
                 float* __restrict__ out) {
    __shared__ float sfs[256], sfc[256], sbs[256], sbc[256];
    float fs = 0.f, fc = 0.f, bs = 0.f, bc = 0.f;
    for (int i = threadIdx.x; i < N_POS; i += 256) {
        float w = rowAny[i] ? 1.f : 0.f;
        fs += sqrtf(fmaxf(minPos[i], 0.f)) * w;
        fc += w;
    }
    for (int i = threadIdx.x; i < N_ANC; i += 256) {
        float w = colAny[i] ? 1.f : 0.f;
        bs += sqrtf(fmaxf(minAnc[i], 0.f)) * w;
        bc += w;
    }
    sfs[threadIdx.x] = fs; sfc[threadIdx.x] = fc;
    sbs[threadIdx.x] = bs; sbc[threadIdx.x] = bc;
    __syncthreads();
    for (int off = 128; off > 0; off >>= 1) {
        if (threadIdx.x < off) {
            sfs[threadIdx.x] += sfs[threadIdx.x + off];
            sfc[threadIdx.x] += sfc[threadIdx.x + off];
            sbs[threadIdx.x] += sbs[threadIdx.x + off];
            sbc[threadIdx.x] += sbc[threadIdx.x + off];
        }
        __syncthreads();
    }
    if (threadIdx.x == 0) out[0] = sfs[0] / sfc[0] + sbs[0] / sbc[0];
}

extern "C" void kernel_launch(void* const* d_in, const int* in_sizes, int n_in,
                              void* d_out, int out_size, void* d_ws, size_t ws_size,
                              hipStream_t stream) {
    (void)in_sizes; (void)n_in; (void)out_size; (void)ws_size;
    const float* pos = (const float*)d_in[0];
    const float* anc = (const float*)d_in[1];
    const float* T   = (const float*)d_in[2];
    const int*  mask = (const int*)d_in[3];

    char* ws = (char*)d_ws;
    float4* posA4 = (float4*)(ws + 0);        // 128 KB
    float4* posB4 = (float4*)(ws + 131072);   // 128 KB
    float4* ancA4 = (float4*)(ws + 262144);   // 128 KB
    float4* ancB4 = (float4*)(ws + 393216);   // 128 KB
    float*  sqPos = (float*) (ws + 524288);   // 32 KB
    float*  sqAnc = (float*) (ws + 557056);   // 32 KB
    float*  minPos= (float*) (ws + 589824);   // 32 KB
    float*  minAnc= (float*) (ws + 622592);   // 32 KB
    int*    rowAny= (int*)   (ws + 655360);   // 32 KB
    int*    colAny= (int*)   (ws + 688128);   // 32 KB

    prep_kernel<<<32, 256, 0, stream>>>(pos, anc, T, posA4, posB4, sqPos, ancA4, ancB4, sqAnc);
    zero_kernel<<<32, 256, 0, stream>>>(colAny, N_ANC);
    // forward: for each pos row, min over all anchors
    chamfer_min_kernel<<<64, 256, 0, stream>>>(posA4, sqPos, ancB4, minPos, N_ANC);
    // backward: for each anchor row, min over all pos (d2 is symmetric in roles)
    chamfer_min_kernel<<<64, 256, 0, stream>>>(ancA4, sqAnc, posB4, minAnc, N_POS);
    mask_kernel<<<64, 256, 0, stream>>>(mask, rowAny, colAny);
    loss_kernel<<<1, 256, 0, stream>>>(minPos, rowAny, minAnc, colAny, (float*)d_out);
}